// medt_net_41205916238509
// MI455X (gfx1250) — compile-verified
//
#include <hip/hip_runtime.h>
#include <hip/hip_bf16.h>
#include <math.h>

// ---------------------------------------------------------------------------
// Deformable conv block on MI455X (gfx1250).
//   out[b,oc,p] = sum_k W[oc,k] * V[b,k,p],  k = ic*9 + n, p = h*128+w
// V = modulated bilinear-sampled x. Final GEMM uses V_WMMA_F32_16X16X4_F32
// (exact fp32, wave32), M-blocked 4x: each wave computes a 64(oc) x 16(p)
// tile so every B fragment feeds 4 WMMAs.
// ---------------------------------------------------------------------------

typedef float v2f __attribute__((ext_vector_type(2)));
typedef float v8f __attribute__((ext_vector_type(8)));

#define BATCH   4
#define INC     64
#define OUTC    128
#define HW      128
#define NPIX    (HW * HW)          // 16384
#define KDIM    (INC * 9)          // 576
#define KTILES  (KDIM / 4)         // 144
#define PTILES  (NPIX / 16)        // 1024
#define OCTILES (OUTC / 16)        // 8
#define HP      130                // padded spatial
#define OCSTRIDE (KTILES * 32 * 2) // floats per octile in Apack = 9216

__device__ __forceinline__ float sigmoidf_(float v) {
    return 1.0f / (1.0f + __expf(-v));
}

// ---------------------------------------------------------------------------
// K1: per-pixel small convs (offset 18ch, mask 9ch, ad 3ch) + deformable
// sampling geometry. Emits 4 (index, weight) pairs per (b, n, p); the
// modulation mask m[n] is folded into the weights; out-of-image corners get
// weight 0 (the padded ring of x is zero).
// Metadata layout: idx4/wgt4[((b*9 + n)*NPIX + p)*4 + c]  -> one int4/float4
// per (b,n,p), 16B aligned, lane-coalesced in p.
// One thread per (b, h, w). 65536 threads.
// ---------------------------------------------------------------------------
__global__ void k_prep(const float* __restrict__ x,
                       const float* __restrict__ w_p, const float* __restrict__ b_p,
                       const float* __restrict__ w_m, const float* __restrict__ b_m,
                       const float* __restrict__ w_ad, const float* __restrict__ b_ad,
                       int* __restrict__ idx4, float* __restrict__ wgt4) {
    int gid = blockIdx.x * blockDim.x + threadIdx.x;           // 0 .. B*NPIX-1
    int b = gid / NPIX;
    int p = gid - b * NPIX;
    int h = p >> 7;
    int w = p & 127;

    float accp[18], accm[9], acca[3];
#pragma unroll
    for (int c = 0; c < 18; ++c) accp[c] = b_p[c];
#pragma unroll
    for (int c = 0; c < 9; ++c) accm[c] = b_m[c];
#pragma unroll
    for (int c = 0; c < 3; ++c) acca[c] = b_ad[c];

    for (int ic = 0; ic < INC; ++ic) {
        const float* xc = x + ((size_t)(b * INC + ic)) * NPIX;
#pragma unroll
        for (int di = 0; di < 3; ++di) {
            int hh = h + di - 1;
            if (hh < 0 || hh > 127) continue;
#pragma unroll
            for (int dj = 0; dj < 3; ++dj) {
                int ww = w + dj - 1;
                if (ww < 0 || ww > 127) continue;
                float xv = xc[hh * HW + ww];
                int tap = ic * 9 + di * 3 + dj;
#pragma unroll
                for (int c = 0; c < 18; ++c)
                    accp[c] = fmaf(w_p[c * KDIM + tap], xv, accp[c]);
#pragma unroll
                for (int c = 0; c < 9; ++c)
                    accm[c] = fmaf(w_m[c * KDIM + tap], xv, accm[c]);
#pragma unroll
                for (int c = 0; c < 3; ++c)
                    acca[c] = fmaf(w_ad[c * KDIM + tap], xv, acca[c]);
            }
        }
    }

    float ad3[3], m9[9];
#pragma unroll
    for (int c = 0; c < 3; ++c) ad3[c] = 2.0f * (1.0f - sigmoidf_(acca[c]));
#pragma unroll
    for (int c = 0; c < 9; ++c) m9[c] = sigmoidf_(accm[c]);

#pragma unroll
    for (int n = 0; n < 9; ++n) {
        int i = n / 3, j = n % 3;
        float pnx = (float)(i - 1);
        float pny = (float)(j - 1);
        float adv = ad3[n % 3];
        // position in PADDED coords (pad=1): p0 = h+1 / w+1
        float px = (float)(h + 1) + pnx + accp[n] + adv * pnx;
        float py = (float)(w + 1) + pny + accp[9 + n] + adv * pny;

        float fltx = floorf(px), flty = floorf(py);
        // clipped integer corners in [0, HP-1]
        float cltx = fminf(fmaxf(fltx, 0.0f), (float)(HP - 1));
        float clty = fminf(fmaxf(flty, 0.0f), (float)(HP - 1));
        float crbx = fminf(fmaxf(fltx + 1.0f, 0.0f), (float)(HP - 1));
        float crby = fminf(fmaxf(flty + 1.0f, 0.0f), (float)(HP - 1));

        // mask + snap + clip of the continuous position (reference semantics)
        bool mx = (px < 1.0f) || (px > (float)(HP - 2));
        bool my = (py < 1.0f) || (py > (float)(HP - 2));
        float pxa = mx ? fltx : px;
        float pya = my ? flty : py;
        pxa = fminf(fmaxf(pxa, 0.0f), (float)(HP - 1));
        pya = fminf(fmaxf(pya, 0.0f), (float)(HP - 1));

        float g_lt = (1.0f + (cltx - pxa)) * (1.0f + (clty - pya));
        float g_rb = (1.0f - (crbx - pxa)) * (1.0f - (crby - pya));
        float g_lb = (1.0f + (cltx - pxa)) * (1.0f - (crby - pya));
        float g_rt = (1.0f - (crbx - pxa)) * (1.0f + (clty - pya));

        int qx[4] = {(int)cltx, (int)crbx, (int)cltx, (int)crbx};
        int qy[4] = {(int)clty, (int)crby, (int)crby, (int)clty};
        float gg[4] = {g_lt, g_rb, g_lb, g_rt};

        int4  iv;
        float4 wv;
        int* ivp = &iv.x;
        float* wvp = &wv.x;
#pragma unroll
        for (int c = 0; c < 4; ++c) {
            int ux = qx[c] - 1, uy = qy[c] - 1;     // unpadded coords
            bool valid = (ux >= 0) && (ux < HW) && (uy >= 0) && (uy < HW);
            ivp[c] = valid ? (ux * HW + uy) : 0;
            wvp[c] = valid ? (gg[c] * m9[n]) : 0.0f;
        }
        size_t mbase = ((size_t)(b * 9 + n) * NPIX + p) * 4;
        *(int4*)(idx4 + mbase) = iv;
        *(float4*)(wgt4 + mbase) = wv;
    }
}

// ---------------------------------------------------------------------------
// K2: build V[b][ic*9+n][p].  One thread per (b, n, p); metadata loaded ONCE
// (aligned int4+float4, lane-coalesced), then loop over all 64 input
// channels: 4 gathers + blend + coalesced store per channel.
// ---------------------------------------------------------------------------
__global__ void k_sample(const float* __restrict__ x,
                         const int* __restrict__ idx4,
                         const float* __restrict__ wgt4,
                         float* __restrict__ V) {
    int gid = blockIdx.x * blockDim.x + threadIdx.x;   // 0 .. B*9*NPIX-1
    int b = gid / (9 * NPIX);
    int rem = gid - b * (9 * NPIX);
    int n = rem >> 14;            // / NPIX
    int p = rem & (NPIX - 1);

    size_t mbase = (size_t)gid * 4;
    int4   iv = *(const int4*)(idx4 + mbase);
    float4 wv = *(const float4*)(wgt4 + mbase);

    const float* xs = x + (size_t)b * INC * NPIX;      // channel 0 of batch b
    float* vp = V + (size_t)b * (KDIM * NPIX) + (size_t)n * NPIX + p;

#pragma unroll 4
    for (int ic = 0; ic < INC; ++ic) {
        const float* xc = xs + (size_t)ic * NPIX;
        float s;
        s = wv.x * xc[iv.x];
        s = fmaf(wv.y, xc[iv.y], s);
        s = fmaf(wv.z, xc[iv.z], s);
        s = fmaf(wv.w, xc[iv.w], s);
        vp[(size_t)ic * 9 * NPIX] = s;
    }
}

// ---------------------------------------------------------------------------
// K3: pack w_conv[128][576] into WMMA f32 16x16x4 A-fragment order:
//   Apack[((octile*KTILES + kt)*32 + lane)*2 + r]
//     = W[octile*16 + (lane&15)][4*kt + ((lane>>4)*2) + r]
// (A 16x4: lanes 0-15 hold M rows with K={0,1}; lanes 16-31 same rows K={2,3})
// ---------------------------------------------------------------------------
__global__ void k_packA(const float* __restrict__ w_conv,
                        float* __restrict__ Apack) {
    int t = blockIdx.x * blockDim.x + threadIdx.x;     // 0 .. OCTILES*KTILES*32-1
    int lane = t & 31;
    int kt = (t >> 5) % KTILES;
    int octile = t / (KTILES * 32);
    int oc = octile * 16 + (lane & 15);
    int kb = 4 * kt + ((lane >> 4) << 1);
    Apack[t * 2 + 0] = w_conv[oc * KDIM + kb + 0];
    Apack[t * 2 + 1] = w_conv[oc * KDIM + kb + 1];
}

// ---------------------------------------------------------------------------
// K4: GEMM out[b][oc][p] = sum_k W[oc][k] * V[b][k][p], M-blocked 4x.
// One wave computes 4 oc-tiles (64 rows) x 16 cols; each B fragment is used
// by 4 v_wmma_f32_16x16x4_f32.  Prefetch the streaming V operand 2 k-steps
// ahead (global_prefetch_b8; speculative, safely dropped if OOB).
// B fragment: VGPR r, lane l -> B[(l>=16?2:0)+r][l&15].
// C/D: VGPR v, lane l -> M = v + 8*(l>=16), N = l&15.
// ---------------------------------------------------------------------------
__global__ void k_gemm(const float* __restrict__ Apack,
                       const float* __restrict__ V,
                       float* __restrict__ out) {
    int lane = threadIdx.x & 31;
    int wave = threadIdx.x >> 5;
    int tile = blockIdx.x * 8 + wave;          // 0 .. B*2*PTILES-1
    int ptile = tile & (PTILES - 1);
    int og    = (tile >> 10) & 1;              // oc-group: octiles og*4 .. og*4+3
    int b     = tile >> 11;

    int col = (ptile << 4) + (lane & 15);
    int krlane = (lane >> 4) << 1;             // 0 or 2

    const float* Ap0 = Apack + ((size_t)(og * 4) * KTILES * 32 + lane) * 2;
    const float* Vb  = V + (size_t)b * (KDIM * NPIX) + col;

    v8f acc0 = {}, acc1 = {}, acc2 = {}, acc3 = {};
#pragma unroll 2
    for (int kt = 0; kt < KTILES; ++kt) {
        const float* ap = Ap0 + (size_t)kt * 64;
        v2f a0 = *(const v2f*)(ap);
        v2f a1 = *(const v2f*)(ap + OCSTRIDE);
        v2f a2 = *(const v2f*)(ap + 2 * OCSTRIDE);
        v2f a3 = *(const v2f*)(ap + 3 * OCSTRIDE);
        const float* vp = Vb + (size_t)(4 * kt + krlane) * NPIX;
        v2f bb;
        bb.x = vp[0];
        bb.y = vp[NPIX];
        __builtin_prefetch(vp + 8 * NPIX, 0, 1);   // stream V 2 k-steps ahead
        acc0 = __builtin_amdgcn_wmma_f32_16x16x4_f32(false, a0, false, bb,
                                                     (short)0, acc0, false, false);
        acc1 = __builtin_amdgcn_wmma_f32_16x16x4_f32(false, a1, false, bb,
                                                     (short)0, acc1, false, false);
        acc2 = __builtin_amdgcn_wmma_f32_16x16x4_f32(false, a2, false, bb,
                                                     (short)0, acc2, false, false);
        acc3 = __builtin_amdgcn_wmma_f32_16x16x4_f32(false, a3, false, bb,
                                                     (short)0, acc3, false, false);
    }

    float* op = out + (size_t)(b * OUTC) * NPIX + col;
    int rb = (og << 6) + ((lane >> 4) << 3);   // first M row held by this lane
    v8f accs[4] = {acc0, acc1, acc2, acc3};
#pragma unroll
    for (int i = 0; i < 4; ++i) {
#pragma unroll
        for (int v = 0; v < 8; ++v)
            op[(size_t)(rb + i * 16 + v) * NPIX] = accs[i][v];
    }
}

// ---------------------------------------------------------------------------
extern "C" void kernel_launch(void* const* d_in, const int* in_sizes, int n_in,
                              void* d_out, int out_size, void* d_ws, size_t ws_size,
                              hipStream_t stream) {
    const float* x      = (const float*)d_in[0];
    const float* w_p    = (const float*)d_in[1];
    const float* b_p    = (const float*)d_in[2];
    const float* w_m    = (const float*)d_in[3];
    const float* b_m    = (const float*)d_in[4];
    const float* w_ad   = (const float*)d_in[5];
    const float* b_ad   = (const float*)d_in[6];
    const float* w_conv = (const float*)d_in[7];
    float* out = (float*)d_out;

    // workspace layout (~170 MB)
    const size_t vElems   = (size_t)BATCH * KDIM * NPIX;      // 37,748,736
    const size_t pairElem = (size_t)BATCH * 9 * NPIX * 4;     //  2,359,296
    float* V     = (float*)d_ws;
    int*   idx4  = (int*)(V + vElems);
    float* wgt4  = (float*)(idx4 + pairElem);
    float* Apack = (float*)(wgt4 + pairElem);                 // 73,728 floats

    // K1: sampling geometry + small convs.  B*NPIX = 65536 threads.
    k_prep<<<(BATCH * NPIX) / 256, 256, 0, stream>>>(
        x, w_p, b_p, w_m, b_m, w_ad, b_ad, idx4, wgt4);

    // K3: pack A fragments (independent of K1).
    k_packA<<<(OCTILES * KTILES * 32) / 256, 256, 0, stream>>>(w_conv, Apack);

    // K2: gather/blend into V.  B*9*NPIX = 589,824 threads, 64 ch each.
    k_sample<<<(BATCH * 9 * NPIX) / 256, 256, 0, stream>>>(x, idx4, wgt4, V);

    // K4: WMMA GEMM.  B*2*PTILES = 8192 waves, 8 waves/block.
    k_gemm<<<(BATCH * 2 * PTILES) / 8, 256, 0, stream>>>(Apack, V, out);
}